// GraphConv_39058432590090
// MI455X (gfx1250) — compile-verified
//
#include <hip/hip_runtime.h>

typedef __attribute__((ext_vector_type(2))) float v2f;
typedef __attribute__((ext_vector_type(8))) float v8f;

#define N_NODES 100000
#define N_EDGES 1600000
#define D       128   // D_IN == D_OUT == 128

// ---------------------------------------------------------------------------
// Kernel 1: zero the aggregation buffer (d_out doubles as agg scratch).
// ---------------------------------------------------------------------------
__global__ void gc_zero_kernel(float4* __restrict__ out, int n4) {
    int i = blockIdx.x * blockDim.x + threadIdx.x;
    if (i < n4) out[i] = make_float4(0.f, 0.f, 0.f, 0.f);
}

// ---------------------------------------------------------------------------
// Kernel 2: SpMM via f32 global atomics. One wave per edge: 32 lanes cover the
// 128-float feature row as float4 chunks (coalesced 512B row gather of x[col],
// which is L2-resident on MI455X's 192MB L2), then 4 global_atomic_add_f32
// into agg[row] (also L2-resident -> atomics resolve in L2).
// ---------------------------------------------------------------------------
__global__ void __launch_bounds__(256)
gc_spmm_kernel(const float* __restrict__ x,
               const int*   __restrict__ erow,
               const int*   __restrict__ ecol,
               const float* __restrict__ eval,
               float*       __restrict__ agg) {
    int g = blockIdx.x * 256 + threadIdx.x;      // E*32 = 51.2M threads total
    int e = g >> 5;                              // edge index
    int f = (g & 31) * 4;                        // feature chunk [f, f+3]
    if (e >= N_EDGES) return;

    int   r = erow[e];                           // broadcast within wave
    int   c = ecol[e];
    float v = eval[e];

    const float4 xv = *(const float4*)(x + (long long)c * D + f);
    float* dst = agg + (long long)r * D + f;
    atomicAdd(dst + 0, v * xv.x);
    atomicAdd(dst + 1, v * xv.y);
    atomicAdd(dst + 2, v * xv.z);
    atomicAdd(dst + 3, v * xv.w);
}

// ---------------------------------------------------------------------------
// Kernel 3: out = agg @ W^T + bias, in place, using V_WMMA_F32_16X16X4_F32.
//
// Per block: 256 threads = 8 waves. W (128x128 f32 = 64KB) staged into LDS
// with an XOR-8 swizzle inside each 128-float row so the 16 lanes of a B-
// fragment read (stride 512B apart) land on spread banks.
//
// Per wave: one 16(M)x128(N) tile. 8 v8f accumulators (one per 16-col N-tile)
// seeded with bias. K-loop: 32 steps of K=4; per step one global b64 A-frag
// load and 8 LDS b64 B-frag loads feeding 8 WMMAs.
//
// Fragment layout (ISA 7.12.2, 32-bit A 16x4 / B 4x16):
//   lanes 0-15 : v0=K(k0+0), v1=K(k0+1); lanes 16-31: v0=K(k0+2), v1=K(k0+3)
//   A row m = m0 + lane%16 ; B col n = n0 + lane%16, B[k][n] = W[n][k].
// C/D: VGPR r, lane l -> row m0 + r + 8*(l/16), col n0 + l%16.
// In-place is safe: all K reads of rows m0..m0+15 happen before any store,
// and rows are disjoint across waves/blocks.
// ---------------------------------------------------------------------------
__global__ void __launch_bounds__(256)
gc_gemm_kernel(float* __restrict__ inout,        // [N_NODES][D] agg -> out
               const float* __restrict__ W,      // [D_OUT][D_IN] row-major
               const float* __restrict__ bias) { // [D_OUT]
    __shared__ float wl[D * D];                  // exactly 64 KB

    const int tid = threadIdx.x;

    // Stage W into LDS (coalesced float4 loads, XOR-8 swizzled placement).
    for (int i = tid * 4; i < D * D; i += 256 * 4) {
        int n = i / D;
        int k = i % D;                           // multiple of 4
        float4 wv = *(const float4*)(W + i);
        int kk = k ^ ((n & 15) * 8);             // flips bits >=3 only
        *(float4*)(&wl[n * D + kk]) = wv;
    }
    __syncthreads();

    const int wave    = tid >> 5;                // 0..7
    const int lane    = tid & 31;
    const int lane16  = lane & 15;
    const int halfsel = lane >> 4;               // 0 or 1

    const long long mtile = (long long)blockIdx.x * 8 + wave;
    if (mtile * 16 >= N_NODES) return;           // after barrier: safe
    const long long m0 = mtile * 16;

    // Per-lane A base: row m0+lane16, K offset 2*halfsel within each k-step.
    const float* arow = inout + (m0 + lane16) * D + 2 * halfsel;

    // Accumulators seeded with bias (bias depends only on column).
    v8f acc[8];
#pragma unroll
    for (int nt = 0; nt < 8; ++nt) {
        float b = bias[nt * 16 + lane16];
#pragma unroll
        for (int r = 0; r < 8; ++r) acc[nt][r] = b;
    }

    const int sw = lane16 * 8;                   // LDS bank swizzle
#pragma unroll 2
    for (int k0 = 0; k0 < D; k0 += 4) {
        v2f a = *(const v2f*)(arow + k0);        // global_load_b64
#pragma unroll
        for (int nt = 0; nt < 8; ++nt) {
            int n  = nt * 16 + lane16;
            int kk = (k0 + 2 * halfsel) ^ sw;    // preserves pair adjacency
            v2f b = *(const v2f*)(&wl[n * D + kk]); // ds_load_b64
            acc[nt] = __builtin_amdgcn_wmma_f32_16x16x4_f32(
                false, a, false, b, (short)0, acc[nt], false, false);
        }
    }

    // Store D-matrix fragments (in place over the consumed agg rows).
#pragma unroll
    for (int nt = 0; nt < 8; ++nt) {
#pragma unroll
        for (int r = 0; r < 8; ++r) {
            inout[(m0 + r + 8 * halfsel) * D + nt * 16 + lane16] = acc[nt][r];
        }
    }
}

// ---------------------------------------------------------------------------
extern "C" void kernel_launch(void* const* d_in, const int* in_sizes, int n_in,
                              void* d_out, int out_size, void* d_ws, size_t ws_size,
                              hipStream_t stream) {
    (void)in_sizes; (void)n_in; (void)out_size; (void)d_ws; (void)ws_size;

    const float* x    = (const float*)d_in[0];
    const int*   erow = (const int*)  d_in[1];
    const int*   ecol = (const int*)  d_in[2];
    const float* eval = (const float*)d_in[3];
    const float* W    = (const float*)d_in[4];
    const float* bias = (const float*)d_in[5];
    float*       out  = (float*)d_out;

    // 1) zero agg (= d_out)
    const int n4 = (N_NODES * D) / 4;            // 3,200,000 float4s
    gc_zero_kernel<<<(n4 + 255) / 256, 256, 0, stream>>>((float4*)out, n4);

    // 2) SpMM with atomics: E * 32 threads
    const long long spmm_threads = (long long)N_EDGES * 32;
    const int spmm_blocks = (int)((spmm_threads + 255) / 256);   // 200,000
    gc_spmm_kernel<<<spmm_blocks, 256, 0, stream>>>(x, erow, ecol, eval, out);

    // 3) in-place WMMA projection + bias
    const int mtiles = (N_NODES + 15) / 16;      // 6250
    const int blocks = (mtiles + 7) / 8;         // 782
    gc_gemm_kernel<<<blocks, 256, 0, stream>>>(out, W, bias);
}